// AFNOBlock2D_85435489452648
// MI455X (gfx1250) — compile-verified
//
#include <hip/hip_runtime.h>
#include <cmath>

// ---------------------------------------------------------------------------
// AFNO block on MI455X (gfx1250), fp32 end-to-end.
// Every dense op (row/col DFTs, block-diagonal complex MLP, pixel MLP) is
// expressed as 16x16x4 fp32 WMMA GEMMs. Total ~250 GFLOP fp32 vs a ~56us
// HBM floor (1.3 GB traffic @ 23.3 TB/s) -> bandwidth bound, so fp32 WMMA
// (exact reference numerics) costs nothing vs bf16 on this workload.
// Ragged DFT edges (129 of 256 rfft bins) are handled by zero-padding the
// twiddle tables / intermediates so the WMMA K-loops carry no mask VALU.
// ---------------------------------------------------------------------------

typedef float v2f __attribute__((ext_vector_type(2)));
typedef float v8f __attribute__((ext_vector_type(8)));

#define DEV __device__ __forceinline__

DEV v8f wmma4(v2f a, v2f b, v8f c) {
  // D = A(16x4) * B(4x16) + C, fp32. neg_a/neg_b unused for f32, c_mod=0.
  return __builtin_amdgcn_wmma_f32_16x16x4_f32(false, a, false, b, (short)0, c,
                                               false, false);
}

DEV v8f v8zero() {
  v8f z = {0.f, 0.f, 0.f, 0.f, 0.f, 0.f, 0.f, 0.f};
  return z;
}

// Problem constants
constexpr long CHW = 16777216;             // 256*256*256 per sample
constexpr int LD1 = 144;                   // padded ld for stage-1 output
constexpr int LDG = 132;                   // padded ld for inverse-row tables
constexpr int FLATN = 129 * 256;           // 33024
constexpr int LD5 = 132 * 256;             // 33792, kf-padded Z5 stride per bh

// Workspace layout (float offsets)
constexpr size_t OFF_FR = 0;                          // [256][144] fwd row cos
constexpr size_t OFF_FI = OFF_FR + 256 * 144;         // [256][144] fwd row -sin
constexpr size_t OFF_ER = OFF_FI + 256 * 144;         // [256][256] fwd col cos
constexpr size_t OFF_EI = OFF_ER + 65536;             // [256][256] fwd col -sin
constexpr size_t OFF_BR = OFF_EI + 65536;             // [256][256] inv col cos
constexpr size_t OFF_BI = OFF_BR + 65536;             // [256][256] inv col +sin
constexpr size_t OFF_GR = OFF_BI + 65536;             // [256][132] inv row
constexpr size_t OFF_GI = OFF_GR + 256 * 132;         // [256][132] inv row
constexpr size_t OFF_ST1 = OFF_GI + 256 * 132;        // mu/rs per sample (4)
constexpr size_t OFF_ST2 = OFF_ST1 + 4;               // mu/rs for x1 (4)
constexpr size_t OFF_PART = OFF_ST2 + 4;              // 2*256*2 partials
constexpr size_t OFF_Z1R = OFF_PART + 1024;
constexpr size_t SZ_Z1 = (size_t)131072 * LD1;        // 18,874,368 floats
constexpr size_t OFF_Z1I = OFF_Z1R + SZ_Z1;
constexpr size_t OFF_Z2R = OFF_Z1I + SZ_Z1;
constexpr size_t SZ_Z2 = (size_t)66048 * 256;         // 16,908,288 floats
constexpr size_t OFF_Z2I = OFF_Z2R + SZ_Z2;
// Z5 ([b,h][kf(132 padded)][c], 512*33792 = 17,301,504 floats) reuses Z1.

// ---------------------------------------------------------------------------
// Twiddle tables (ortho scaling 1/sqrt(256)=1/16 folded per 1-D transform).
// Columns >= 129 of F and G tables are zero (edge handling without masks).
// ---------------------------------------------------------------------------
__global__ void k_init_tables(float* Fr, float* Fi, float* Er, float* Ei,
                              float* Br, float* Bi, float* Gr, float* Gi) {
  const float S = 0.0625f;
  const float W0 = 6.283185307179586f / 256.0f;
  int stride = gridDim.x * blockDim.x;
  int idx0 = blockIdx.x * blockDim.x + threadIdx.x;
  for (int t = idx0; t < 256 * LD1; t += stride) {
    int w = t / LD1, k = t - w * LD1;
    if (k < 129) {
      float ang = W0 * (float)((w * k) & 255);
      float sv, cv;
      sincosf(ang, &sv, &cv);
      Fr[t] = cv * S;            // forward rfft: e^{-i}
      Fi[t] = -sv * S;
    } else {
      Fr[t] = 0.f;
      Fi[t] = 0.f;
    }
  }
  for (int t = idx0; t < 256 * LDG; t += stride) {
    int w = t / LDG, k = t - w * LDG;
    if (k < 129) {
      float ang = W0 * (float)((w * k) & 255);
      float sv, cv;
      sincosf(ang, &sv, &cv);
      float ck = (k == 0 || k == 128) ? 1.0f : 2.0f;   // Hermitian fold
      Gr[t] = ck * cv * S;       // irfft: Re(c_k * Z * e^{+i})
      Gi[t] = -ck * sv * S;
    } else {
      Gr[t] = 0.f;
      Gi[t] = 0.f;
    }
  }
  for (int t = idx0; t < 256 * 256; t += stride) {
    int m = t >> 8, h = t & 255;
    float ang = W0 * (float)((m * h) & 255);
    float sv, cv;
    sincosf(ang, &sv, &cv);
    Er[t] = cv * S;              // forward col fft: e^{-i}
    Ei[t] = -sv * S;
    Br[t] = cv * S;              // inverse col fft: e^{+i} (symmetric matrix)
    Bi[t] = sv * S;
  }
}

// Zero the kf = 129..131 pad rows of Z5 (3*256 floats per (b,h)).
__global__ void k_zero_pad(float* __restrict__ Z5r, float* __restrict__ Z5i) {
  int t = blockIdx.x * blockDim.x + threadIdx.x;
  if (t >= 512 * 3 * 256) return;
  int c = t & 255;
  int r = t >> 8;                // 0..1535
  int bh = r / 3, kf = 129 + (r - bh * 3);
  size_t o = (size_t)bh * LD5 + (size_t)kf * 256 + c;
  Z5r[o] = 0.f;
  Z5i[o] = 0.f;
}

// ---------------------------------------------------------------------------
// Deterministic two-stage mean/var reduction over (C,H,W) per sample.
// ---------------------------------------------------------------------------
__global__ void k_reduce_partial(const float* __restrict__ src,
                                 float* __restrict__ part) {
  int b = blockIdx.x >> 8;
  int blk = blockIdx.x & 255;
  const float* p = src + (size_t)b * CHW + (size_t)blk * 65536;
  float s = 0.f, q = 0.f;
  for (int i = threadIdx.x; i < 65536; i += 256) {
    float v = p[i];
    s += v;
    q += v * v;
  }
  __shared__ float sh[512];
  sh[threadIdx.x] = s;
  sh[threadIdx.x + 256] = q;
  __syncthreads();
  for (int st = 128; st > 0; st >>= 1) {
    if ((int)threadIdx.x < st) {
      sh[threadIdx.x] += sh[threadIdx.x + st];
      sh[threadIdx.x + 256] += sh[threadIdx.x + 256 + st];
    }
    __syncthreads();
  }
  if (threadIdx.x == 0) {
    part[blockIdx.x * 2] = sh[0];
    part[blockIdx.x * 2 + 1] = sh[256];
  }
}

__global__ void k_reduce_final(const float* __restrict__ part,
                               float* __restrict__ stats) {
  int b = blockIdx.x;
  __shared__ float sh[512];
  sh[threadIdx.x] = part[(b * 256 + threadIdx.x) * 2];
  sh[threadIdx.x + 256] = part[(b * 256 + threadIdx.x) * 2 + 1];
  __syncthreads();
  for (int st = 128; st > 0; st >>= 1) {
    if ((int)threadIdx.x < st) {
      sh[threadIdx.x] += sh[threadIdx.x + st];
      sh[threadIdx.x + 256] += sh[threadIdx.x + 256 + st];
    }
    __syncthreads();
  }
  if (threadIdx.x == 0) {
    float inv = 1.0f / (float)CHW;
    float mu = sh[0] * inv;
    float var = sh[256] * inv - mu * mu;
    stats[b * 2] = mu;
    stats[b * 2 + 1] = rsqrtf(var + 1e-5f);
  }
}

// ---------------------------------------------------------------------------
// S1: fused norm1 + row rfft.  Z1[row=(b,c,h)][kf] = sum_w xn[row][w]*F[w][kf]
// F is zero-padded to 144 cols -> no masks; pad cols of Z1 get exact zeros.
// ---------------------------------------------------------------------------
__global__ void __launch_bounds__(256) k_s1_rowdft(
    const float* __restrict__ x, const float* __restrict__ stats,
    const float* __restrict__ n1g, const float* __restrict__ n1b,
    const float* __restrict__ Fr, const float* __restrict__ Fi,
    float* __restrict__ Z1r, float* __restrict__ Z1i) {
  const int lane = threadIdx.x & 31;
  const int half = lane >> 4, l15 = lane & 15;
  long gw = (long)blockIdx.x * 8 + (threadIdx.x >> 5);
  const long TOT = (long)(131072 / 16) * 9;
  if (gw >= TOT) return;
  int nt = (int)(gw % 9);
  int mt = (int)(gw / 9);
  int m0 = mt * 16, n0 = nt * 16;
  int row = m0 + l15;
  int b = row >> 16, c = (row >> 8) & 255;
  float mu = stats[b * 2], rs = stats[b * 2 + 1];
  float sc = rs * n1g[c];
  float of = n1b[c] - mu * sc;
  const float* ar = x + (size_t)row * 256;
  const int kb = half * 2;
  int ncol = n0 + l15;   // < 144 always
  v8f cr = v8zero(), ci = v8zero();
  for (int k0 = 0; k0 < 256; k0 += 4) {
    v2f a;
    a.x = ar[k0 + kb] * sc + of;
    a.y = ar[k0 + kb + 1] * sc + of;
    v2f br, bi;
    br.x = Fr[(k0 + kb) * LD1 + ncol];
    br.y = Fr[(k0 + kb + 1) * LD1 + ncol];
    bi.x = Fi[(k0 + kb) * LD1 + ncol];
    bi.y = Fi[(k0 + kb + 1) * LD1 + ncol];
    cr = wmma4(a, br, cr);
    ci = wmma4(a, bi, ci);
  }
#pragma unroll
  for (int r = 0; r < 8; ++r) {
    int mm = m0 + r + half * 8;
    size_t o = (size_t)mm * LD1 + ncol;
    Z1r[o] = cr[r];    // cols >=129 store exact zeros (B pad was zero)
    Z1i[o] = ci[r];
  }
}

// ---------------------------------------------------------------------------
// S2: forward column DFT (complex GEMM), output transposed to [b,m,kf,c].
// Per (b,c): A = E[m][h] (shared twiddle), B = Z1[(b,c)] [h][kf, zero-padded].
// ---------------------------------------------------------------------------
__global__ void __launch_bounds__(256) k_s2_coldft(
    const float* __restrict__ Z1r, const float* __restrict__ Z1i,
    const float* __restrict__ Er, const float* __restrict__ Ei,
    float* __restrict__ Z2r, float* __restrict__ Z2i) {
  const int lane = threadIdx.x & 31;
  const int half = lane >> 4, l15 = lane & 15;
  long gw = (long)blockIdx.x * 8 + (threadIdx.x >> 5);
  const long TOT = 512L * 16 * 9;
  if (gw >= TOT) return;
  int nt = (int)(gw % 9);
  long t = gw / 9;
  int mt = (int)(t % 16);
  int bc = (int)(t / 16);
  int b = bc >> 8, c = bc & 255;
  const float* S1r = Z1r + (size_t)bc * 256 * LD1;
  const float* S1i = Z1i + (size_t)bc * 256 * LD1;
  int m0 = mt * 16, n0 = nt * 16;
  int arow = m0 + l15;
  const int kb = half * 2;
  int ncol = n0 + l15;   // < 144, pad cols read zeros
  v8f accr = v8zero(), acci = v8zero();
  for (int k0 = 0; k0 < 256; k0 += 4) {
    int kpf = (k0 < 224) ? (k0 + 32) : k0;   // clamped prefetch distance
    __builtin_prefetch(S1r + (size_t)kpf * LD1 + ncol, 0, 1);
    __builtin_prefetch(S1i + (size_t)kpf * LD1 + ncol, 0, 1);
    v2f aR, aI, aIn;
    aR.x = Er[arow * 256 + k0 + kb];
    aR.y = Er[arow * 256 + k0 + kb + 1];
    aI.x = Ei[arow * 256 + k0 + kb];
    aI.y = Ei[arow * 256 + k0 + kb + 1];
    aIn.x = -aI.x;
    aIn.y = -aI.y;
    v2f bR, bI;
    bR.x = S1r[(size_t)(k0 + kb) * LD1 + ncol];
    bR.y = S1r[(size_t)(k0 + kb + 1) * LD1 + ncol];
    bI.x = S1i[(size_t)(k0 + kb) * LD1 + ncol];
    bI.y = S1i[(size_t)(k0 + kb + 1) * LD1 + ncol];
    accr = wmma4(aR, bR, accr);
    accr = wmma4(aIn, bI, accr);   // - Ei * Z1i
    acci = wmma4(aR, bI, acci);
    acci = wmma4(aI, bR, acci);
  }
#pragma unroll
  for (int r = 0; r < 8; ++r) {
    int mm = m0 + r + half * 8;
    if (ncol < 129) {
      size_t o = ((size_t)(b * 256 + mm) * 129 + ncol) * 256 + c;
      Z2r[o] = accr[r];
      Z2i[o] = acci[r];
    }
  }
}

// ---------------------------------------------------------------------------
// S3: block-diagonal complex MLP + relu + softshrink, in-place on Z2.
// One wave = 16 positions x one 32-channel block. LDS round-trips o1 from
// C-fragment layout to A-fragment layout for the second GEMM.
// ---------------------------------------------------------------------------
__global__ void __launch_bounds__(256) k_s3_blockmlp(
    float* __restrict__ Zr, float* __restrict__ Zi,
    const float* __restrict__ w1, const float* __restrict__ b1,
    const float* __restrict__ w2, const float* __restrict__ b2) {
  __shared__ float lds[8][2][16 * 32];
  const int lane = threadIdx.x & 31;
  const int half = lane >> 4, l15 = lane & 15;
  const int wv = threadIdx.x >> 5;
  long gw = (long)blockIdx.x * 8 + wv;   // grid exact: 4128*8 = 33024 waves
  int nb = (int)(gw & 7);
  int pt = (int)(gw >> 3);
  int p = pt * 16 + l15;
  const float* w1r = w1 + (size_t)nb * 1024;
  const float* w1i = w1 + (size_t)(8 + nb) * 1024;
  const float* w2r = w2 + (size_t)nb * 1024;
  const float* w2i = w2 + (size_t)(8 + nb) * 1024;
  const int kb = half * 2;
  size_t abase = (size_t)p * 256 + nb * 32;

  v8f o1r[2], o1i[2];
#pragma unroll
  for (int nh = 0; nh < 2; ++nh) {
    float br_ = b1[nb * 32 + nh * 16 + l15];
    float bi_ = b1[(8 + nb) * 32 + nh * 16 + l15];
#pragma unroll
    for (int r = 0; r < 8; ++r) {
      o1r[nh][r] = br_;
      o1i[nh][r] = bi_;
    }
  }
  for (int k0 = 0; k0 < 32; k0 += 4) {
    v2f axr, axi, axin;
    axr.x = Zr[abase + k0 + kb];
    axr.y = Zr[abase + k0 + kb + 1];
    axi.x = Zi[abase + k0 + kb];
    axi.y = Zi[abase + k0 + kb + 1];
    axin.x = -axi.x;
    axin.y = -axi.y;
#pragma unroll
    for (int nh = 0; nh < 2; ++nh) {
      int o = nh * 16 + l15;
      v2f bwr, bwi;
      bwr.x = w1r[(k0 + kb) * 32 + o];
      bwr.y = w1r[(k0 + kb + 1) * 32 + o];
      bwi.x = w1i[(k0 + kb) * 32 + o];
      bwi.y = w1i[(k0 + kb + 1) * 32 + o];
      o1r[nh] = wmma4(axr, bwr, o1r[nh]);
      o1r[nh] = wmma4(axin, bwi, o1r[nh]);   // - xi*W1i
      o1i[nh] = wmma4(axi, bwr, o1i[nh]);
      o1i[nh] = wmma4(axr, bwi, o1i[nh]);    // + xr*W1i
    }
  }
#pragma unroll
  for (int nh = 0; nh < 2; ++nh)
#pragma unroll
    for (int r = 0; r < 8; ++r) {
      int mm = r + half * 8, nn = nh * 16 + l15;
      lds[wv][0][mm * 32 + nn] = fmaxf(o1r[nh][r], 0.f);
      lds[wv][1][mm * 32 + nn] = fmaxf(o1i[nh][r], 0.f);
    }
  __syncthreads();   // uniform: grid is exact, no early returns

  v8f o2r[2], o2i[2];
#pragma unroll
  for (int nh = 0; nh < 2; ++nh) {
    float br_ = b2[nb * 32 + nh * 16 + l15];
    float bi_ = b2[(8 + nb) * 32 + nh * 16 + l15];
#pragma unroll
    for (int r = 0; r < 8; ++r) {
      o2r[nh][r] = br_;
      o2i[nh][r] = bi_;
    }
  }
  for (int k0 = 0; k0 < 32; k0 += 4) {
    v2f ahr, ahi, ahin;
    ahr.x = lds[wv][0][l15 * 32 + k0 + kb];
    ahr.y = lds[wv][0][l15 * 32 + k0 + kb + 1];
    ahi.x = lds[wv][1][l15 * 32 + k0 + kb];
    ahi.y = lds[wv][1][l15 * 32 + k0 + kb + 1];
    ahin.x = -ahi.x;
    ahin.y = -ahi.y;
#pragma unroll
    for (int nh = 0; nh < 2; ++nh) {
      int o = nh * 16 + l15;
      v2f bwr, bwi;
      bwr.x = w2r[(k0 + kb) * 32 + o];
      bwr.y = w2r[(k0 + kb + 1) * 32 + o];
      bwi.x = w2i[(k0 + kb) * 32 + o];
      bwi.y = w2i[(k0 + kb + 1) * 32 + o];
      o2r[nh] = wmma4(ahr, bwr, o2r[nh]);
      o2r[nh] = wmma4(ahin, bwi, o2r[nh]);
      o2i[nh] = wmma4(ahi, bwr, o2i[nh]);
      o2i[nh] = wmma4(ahr, bwi, o2i[nh]);
    }
  }
#pragma unroll
  for (int nh = 0; nh < 2; ++nh)
#pragma unroll
    for (int r = 0; r < 8; ++r) {
      int mm = r + half * 8, nn = nh * 16 + l15;
      size_t o = (size_t)(pt * 16 + mm) * 256 + nb * 32 + nn;
      float vr = o2r[nh][r];
      float arr = fabsf(vr) - 0.01f;
      Zr[o] = arr > 0.f ? copysignf(arr, vr) : 0.f;
      float vi = o2i[nh][r];
      float aii = fabsf(vi) - 0.01f;
      Zi[o] = aii > 0.f ? copysignf(aii, vi) : 0.f;
    }
}

// ---------------------------------------------------------------------------
// S5: inverse column DFT. Per b: out[h][kf*256+c] = sum_m Binv[h][m]*Z[m][..]
// Output Z5 has kf padded to 132 per (b,h) row block (pad zeroed separately).
// ---------------------------------------------------------------------------
__global__ void __launch_bounds__(256) k_s5_icoldft(
    const float* __restrict__ Z2r, const float* __restrict__ Z2i,
    const float* __restrict__ Br, const float* __restrict__ Bi,
    float* __restrict__ Z5r, float* __restrict__ Z5i) {
  const int lane = threadIdx.x & 31;
  const int half = lane >> 4, l15 = lane & 15;
  long gw = (long)blockIdx.x * 8 + (threadIdx.x >> 5);
  const long TOT = 2L * 16 * 2064;
  if (gw >= TOT) return;
  int nt = (int)(gw % 2064);
  long t = gw / 2064;
  int mt = (int)(t % 16);
  int b = (int)(t / 16);
  const float* Sr = Z2r + (size_t)b * 256 * FLATN;
  const float* Si = Z2i + (size_t)b * 256 * FLATN;
  int m0 = mt * 16, n0 = nt * 16;
  int arow = m0 + l15;
  const int kb = half * 2;
  int ncol = n0 + l15;
  v8f accr = v8zero(), acci = v8zero();
  for (int k0 = 0; k0 < 256; k0 += 4) {
    int kpf = (k0 < 224) ? (k0 + 32) : k0;   // clamped prefetch distance
    __builtin_prefetch(Sr + (size_t)kpf * FLATN + ncol, 0, 1);
    __builtin_prefetch(Si + (size_t)kpf * FLATN + ncol, 0, 1);
    v2f aR, aI, aIn;
    aR.x = Br[arow * 256 + k0 + kb];
    aR.y = Br[arow * 256 + k0 + kb + 1];
    aI.x = Bi[arow * 256 + k0 + kb];
    aI.y = Bi[arow * 256 + k0 + kb + 1];
    aIn.x = -aI.x;
    aIn.y = -aI.y;
    v2f bR, bI;
    bR.x = Sr[(size_t)(k0 + kb) * FLATN + ncol];
    bR.y = Sr[(size_t)(k0 + kb + 1) * FLATN + ncol];
    bI.x = Si[(size_t)(k0 + kb) * FLATN + ncol];
    bI.y = Si[(size_t)(k0 + kb + 1) * FLATN + ncol];
    accr = wmma4(aR, bR, accr);
    accr = wmma4(aIn, bI, accr);
    acci = wmma4(aR, bI, acci);
    acci = wmma4(aI, bR, acci);
  }
#pragma unroll
  for (int r = 0; r < 8; ++r) {
    int hh = m0 + r + half * 8;
    size_t o = (size_t)(b * 256 + hh) * LD5 + ncol;
    Z5r[o] = accr[r];
    Z5i[o] = acci[r];
  }
}

// ---------------------------------------------------------------------------
// S6: inverse row rfft + AFNO residual + layer-scale; writes x1 into d_out.
// G tables padded to 132 cols, Z5 kf-pad zeroed -> unmasked K-loop.
// epilogue: x1 = gamma1[c]*(f + xn) + x     (xn recomputed from x, stats1)
// ---------------------------------------------------------------------------
__global__ void __launch_bounds__(256) k_s6_irowdft(
    const float* __restrict__ Z5r, const float* __restrict__ Z5i,
    const float* __restrict__ Gr, const float* __restrict__ Gi,
    const float* __restrict__ x, const float* __restrict__ stats,
    const float* __restrict__ n1g, const float* __restrict__ n1b,
    const float* __restrict__ gamma1, float* __restrict__ out) {
  const int lane = threadIdx.x & 31;
  const int half = lane >> 4, l15 = lane & 15;
  long gw = (long)blockIdx.x * 8 + (threadIdx.x >> 5);
  const long TOT = 512L * 16 * 16;
  if (gw >= TOT) return;
  int ct = (int)(gw & 15);
  long t = gw >> 4;
  int wt = (int)(t & 15);
  int bh = (int)(t >> 4);
  int b = bh >> 8, hh = bh & 255;
  const float* Sr = Z5r + (size_t)bh * LD5;
  const float* Si = Z5i + (size_t)bh * LD5;
  int w0 = wt * 16, c0 = ct * 16;
  int arow = w0 + l15;   // output w
  const int kb = half * 2;
  int cc = c0 + l15;     // output c (fixed per lane)
  v8f acc = v8zero();
  for (int k0 = 0; k0 < 132; k0 += 4) {
    int ka = k0 + kb;
    v2f aR, aI;
    aR.x = Gr[arow * LDG + ka];
    aR.y = Gr[arow * LDG + ka + 1];
    aI.x = Gi[arow * LDG + ka];
    aI.y = Gi[arow * LDG + ka + 1];
    v2f bR, bI;
    bR.x = Sr[(size_t)ka * 256 + cc];
    bR.y = Sr[(size_t)(ka + 1) * 256 + cc];
    bI.x = Si[(size_t)ka * 256 + cc];
    bI.y = Si[(size_t)(ka + 1) * 256 + cc];
    acc = wmma4(aR, bR, acc);
    acc = wmma4(aI, bI, acc);   // Gi already carries the minus sign
  }
  float mu = stats[b * 2], rs = stats[b * 2 + 1];
  float sc = rs * n1g[cc];
  float of = n1b[cc] - mu * sc;
  float g1 = gamma1[cc];
  size_t base = ((size_t)(b * 256 + cc) * 256 + hh) * 256 + w0 + half * 8;
#pragma unroll
  for (int r = 0; r < 8; ++r) {
    size_t o = base + r;
    float xv = x[o];
    float f = acc[r] + (xv * sc + of);   // irfft output + normed input
    out[o] = g1 * f + xv;                // x1
  }
}

// ---------------------------------------------------------------------------
// S8: fused norm2 + MLP (256->1024 GELU ->256) + layer-scale residual,
// in-place on d_out. One block = 16 pixels; fc1 hidden tiled 8 N-tiles/wave.
// ---------------------------------------------------------------------------
__global__ void __launch_bounds__(256) k_s8_mlp(
    float* __restrict__ out, const float* __restrict__ stats,
    const float* __restrict__ n2g, const float* __restrict__ n2b,
    const float* __restrict__ fc1w, const float* __restrict__ fc1b,
    const float* __restrict__ fc2w, const float* __restrict__ fc2b,
    const float* __restrict__ gamma2) {
  extern __shared__ float sm[];
  float* shA = sm;                 // [16][260] normalized x1
  float* shH = sm + 16 * 260;      // [16][1028] hidden
  const int tid = threadIdx.x;
  const int lane = tid & 31;
  const int half = lane >> 4, l15 = lane & 15;
  const int wv = tid >> 5;
  const long p0 = (long)blockIdx.x * 16;
  const int b = (int)(p0 >> 16);
  const long hw0 = p0 & 65535;
  const float mu = stats[b * 2], rs = stats[b * 2 + 1];

  // Phase A: load + normalize 16-pixel x1 tile (coalesced along pixels)
  for (int i = tid; i < 16 * 256; i += 256) {
    int pr = i & 15, c = i >> 4;
    float v = out[((size_t)(b * 256 + c) << 16) + hw0 + pr];
    shA[pr * 260 + c] = (v - mu) * rs * n2g[c] + n2b[c];
  }
  __syncthreads();

  // Phase B: fc1 + exact-erf GELU; wave wv owns hidden cols [wv*128, +128)
  v8f acc[8];
#pragma unroll
  for (int q = 0; q < 8; ++q) {
    float bb = fc1b[wv * 128 + q * 16 + l15];
#pragma unroll
    for (int r = 0; r < 8; ++r) acc[q][r] = bb;
  }
  {
    const int kb = half * 2;
    for (int k0 = 0; k0 < 256; k0 += 4) {
      v2f a;
      a.x = shA[l15 * 260 + k0 + kb];
      a.y = shA[l15 * 260 + k0 + kb + 1];
#pragma unroll
      for (int q = 0; q < 8; ++q) {
        int o = wv * 128 + q * 16 + l15;
        v2f bw;
        bw.x = fc1w[(size_t)o * 256 + k0 + kb];
        bw.y = fc1w[(size_t)o * 256 + k0 + kb + 1];
        acc[q] = wmma4(a, bw, acc[q]);
      }
    }
  }
#pragma unroll
  for (int q = 0; q < 8; ++q)
#pragma unroll
    for (int r = 0; r < 8; ++r) {
      float v = acc[q][r];
      float gel = 0.5f * v * (1.0f + erff(v * 0.70710678118654752f));
      int mm = r + half * 8, nn = wv * 128 + q * 16 + l15;
      shH[mm * 1028 + nn] = gel;
    }
  __syncthreads();

  // Phase C: fc2 + gamma2 + residual; wave wv owns out cols [wv*32, +32)
  v8f acc2[2];
#pragma unroll
  for (int q = 0; q < 2; ++q) {
    float bb = fc2b[wv * 32 + q * 16 + l15];
#pragma unroll
    for (int r = 0; r < 8; ++r) acc2[q][r] = bb;
  }
  {
    const int kb = half * 2;
    for (int k0 = 0; k0 < 1024; k0 += 4) {
      v2f a;
      a.x = shH[l15 * 1028 + k0 + kb];
      a.y = shH[l15 * 1028 + k0 + kb + 1];
#pragma unroll
      for (int q = 0; q < 2; ++q) {
        int o = wv * 32 + q * 16 + l15;
        v2f bw;
        bw.x = fc2w[(size_t)o * 1024 + k0 + kb];
        bw.y = fc2w[(size_t)o * 1024 + k0 + kb + 1];
        acc2[q] = wmma4(a, bw, acc2[q]);
      }
    }
  }
#pragma unroll
  for (int q = 0; q < 2; ++q) {
    int oc = wv * 32 + q * 16 + l15;
    float g2 = gamma2[oc];
#pragma unroll
    for (int r = 0; r < 8; ++r) {
      int pr = r + half * 8;
      size_t o = ((size_t)(b * 256 + oc) << 16) + hw0 + pr;
      float x1v = out[o];          // raw x1, only this wave touches (p,oc)
      out[o] = g2 * acc2[q][r] + x1v;
    }
  }
}

// ---------------------------------------------------------------------------
extern "C" void kernel_launch(void* const* d_in, const int* in_sizes, int n_in,
                              void* d_out, int out_size, void* d_ws,
                              size_t ws_size, hipStream_t stream) {
  (void)in_sizes; (void)n_in; (void)out_size; (void)ws_size;
  const float* x     = (const float*)d_in[0];
  const float* w1    = (const float*)d_in[1];
  const float* b1    = (const float*)d_in[2];
  const float* w2    = (const float*)d_in[3];
  const float* b2    = (const float*)d_in[4];
  const float* n1g   = (const float*)d_in[5];
  const float* n1b   = (const float*)d_in[6];
  const float* n2g   = (const float*)d_in[7];
  const float* n2b   = (const float*)d_in[8];
  const float* gam1  = (const float*)d_in[9];
  const float* gam2  = (const float*)d_in[10];
  const float* fc1w  = (const float*)d_in[11];
  const float* fc1b  = (const float*)d_in[12];
  const float* fc2w  = (const float*)d_in[13];
  const float* fc2b  = (const float*)d_in[14];
  float* out = (float*)d_out;
  float* ws = (float*)d_ws;

  float* Fr = ws + OFF_FR;  float* Fi = ws + OFF_FI;
  float* Er = ws + OFF_ER;  float* Ei = ws + OFF_EI;
  float* Br = ws + OFF_BR;  float* Bi = ws + OFF_BI;
  float* Gr = ws + OFF_GR;  float* Gi = ws + OFF_GI;
  float* st1 = ws + OFF_ST1;
  float* st2 = ws + OFF_ST2;
  float* part = ws + OFF_PART;
  float* Z1r = ws + OFF_Z1R;  float* Z1i = ws + OFF_Z1I;
  float* Z2r = ws + OFF_Z2R;  float* Z2i = ws + OFF_Z2I;
  float* Z5r = Z1r;  float* Z5i = Z1i;   // reuse (Z5 fits inside Z1)

  k_init_tables<<<256, 256, 0, stream>>>(Fr, Fi, Er, Ei, Br, Bi, Gr, Gi);

  k_reduce_partial<<<512, 256, 0, stream>>>(x, part);
  k_reduce_final<<<2, 256, 0, stream>>>(part, st1);

  k_s1_rowdft<<<9216, 256, 0, stream>>>(x, st1, n1g, n1b, Fr, Fi, Z1r, Z1i);
  k_s2_coldft<<<9216, 256, 0, stream>>>(Z1r, Z1i, Er, Ei, Z2r, Z2i);
  k_s3_blockmlp<<<4128, 256, 0, stream>>>(Z2r, Z2i, w1, b1, w2, b2);
  k_s5_icoldft<<<8256, 256, 0, stream>>>(Z2r, Z2i, Br, Bi, Z5r, Z5i);
  k_zero_pad<<<1536, 256, 0, stream>>>(Z5r, Z5i);
  k_s6_irowdft<<<16384, 256, 0, stream>>>(Z5r, Z5i, Gr, Gi, x, st1, n1g, n1b,
                                          gam1, out);

  k_reduce_partial<<<512, 256, 0, stream>>>(out, part);
  k_reduce_final<<<2, 256, 0, stream>>>(part, st2);

  size_t shbytes = (size_t)(16 * 260 + 16 * 1028) * sizeof(float);
  k_s8_mlp<<<8192, 256, shbytes, stream>>>(out, st2, n2g, n2b, fc1w, fc1b,
                                           fc2w, fc2b, gam2);
}